// Graft_StackedDense_9689446220148
// MI455X (gfx1250) — compile-verified
//
#include <hip/hip_runtime.h>
#include <hip/hip_bf16.h>
#include <math.h>

typedef __attribute__((ext_vector_type(2))) float v2f;
typedef __attribute__((ext_vector_type(4))) float v4f;
typedef __attribute__((ext_vector_type(8))) float v8f;

#define TK 32
#define LDSP (TK + 4)   // 36-float row stride: conflict-free strided column access

// pnet GEMM tile (synchronous staging)
#define TM 128
#define TN 64
// main GEMM tile (async double-buffered staging)
#define TM2 128
#define TN2 128

// ---------------- gfx1250 async global->LDS (ASYNCcnt-tracked) ---------------
__device__ __forceinline__ void async_ld_b128(unsigned lds_addr, const void* sbase,
                                              unsigned voff) {
    // GVS mode: mem = SGPR base + 32-bit per-lane offset; LDS dest addr per lane.
    asm volatile("global_load_async_to_lds_b128 %0, %1, %2"
                 :: "v"(lds_addr), "v"(voff), "s"(sbase)
                 : "memory");
}
#define S_WAIT_ASYNCCNT_8() asm volatile("s_wait_asynccnt 8" ::: "memory")
#define S_WAIT_ASYNCCNT_0() asm volatile("s_wait_asynccnt 0" ::: "memory")

// ---------------- float <-> ordered-uint encoding for atomic min/max ---------
__device__ __forceinline__ unsigned fenc(float f) {
    unsigned u = __float_as_uint(f);
    return (u & 0x80000000u) ? ~u : (u | 0x80000000u);
}
__device__ __forceinline__ float fdec(unsigned v) {
    unsigned u = (v & 0x80000000u) ? (v ^ 0x80000000u) : ~v;
    return __uint_as_float(u);
}

// ---------------- reference-faithful half-open binning -----------------------
__device__ __forceinline__ void bin_add(int* h, float x, float lo, float sc) {
    if (sc > 0.f) {
        int b = (int)((x - lo) / sc);
        if (b < 0) b = 0;
        if (b > 9) b = 9;
        while (b < 10 && x >= lo + (float)(b + 1) * sc) b++;   // max element -> dropped
        while (b > 0  && x <  lo + (float)b * sc)       b--;
        if (b < 10 && x >= lo + (float)b * sc && x < lo + (float)(b + 1) * sc)
            atomicAdd(&h[b], 1);
    }
}

__device__ __forceinline__ float entropy10(const int* h, float n) {
    float H = 0.f;
    for (int i = 0; i < 10; ++i) {
        float p = (float)h[i] / n;
        if (p > 0.f) H -= p * logf(p);
    }
    return H;
}

__device__ __forceinline__ float blend1(float base, float graft, float d, float wg) {
    float wl = 1.f / (1.f + expf(-d));                       // sigmoid
    float wb = wg * (1.f - expf(-wg * wl));
    float wr = (1.f - wg) * (1.f - expf(-(1.f - wg) * (1.f - wl)));
    float e0 = expf(wb), e1 = expf(wr);
    float inv = 1.f / (e0 + e1);
    return base * (e0 * inv) + graft * (e1 * inv);
}

// ---------------- small utility kernels --------------------------------------
__global__ void init_stats_kernel(unsigned* enc, int* hist) {
    int t = threadIdx.x;
    if (t < 2) { enc[2 * t] = 0xFFFFFFFFu; enc[2 * t + 1] = 0u; }
    if (t < 20) hist[t] = 0;
}

__global__ __launch_bounds__(256) void minmax2_kernel(const float* __restrict__ a,
                                                      const float* __restrict__ b,
                                                      int n, unsigned* enc) {
    unsigned amin = 0xFFFFFFFFu, amax = 0u, bmin = 0xFFFFFFFFu, bmax = 0u;
    for (int i = blockIdx.x * blockDim.x + threadIdx.x; i < n; i += gridDim.x * blockDim.x) {
        unsigned ea = fenc(a[i]); amin = min(amin, ea); amax = max(amax, ea);
        unsigned eb = fenc(b[i]); bmin = min(bmin, eb); bmax = max(bmax, eb);
    }
    __shared__ unsigned s[4][256];
    int t = threadIdx.x;
    s[0][t] = amin; s[1][t] = amax; s[2][t] = bmin; s[3][t] = bmax;
    __syncthreads();
    for (int off = 128; off > 0; off >>= 1) {
        if (t < off) {
            s[0][t] = min(s[0][t], s[0][t + off]);
            s[1][t] = max(s[1][t], s[1][t + off]);
            s[2][t] = min(s[2][t], s[2][t + off]);
            s[3][t] = max(s[3][t], s[3][t + off]);
        }
        __syncthreads();
    }
    if (t == 0) {
        atomicMin(&enc[0], s[0][0]); atomicMax(&enc[1], s[1][0]);
        atomicMin(&enc[2], s[2][0]); atomicMax(&enc[3], s[3][0]);
    }
}

__global__ __launch_bounds__(256) void hist_kernel(const float* __restrict__ a, int n,
                                                   const unsigned* __restrict__ enc,
                                                   int* __restrict__ hist) {
    __shared__ int sh[10];
    if (threadIdx.x < 10) sh[threadIdx.x] = 0;
    __syncthreads();
    float lo = fdec(enc[0]), hi = fdec(enc[1]);
    float sc = (hi - lo) / 10.f;
    for (int i = blockIdx.x * blockDim.x + threadIdx.x; i < n; i += gridDim.x * blockDim.x)
        bin_add(sh, a[i], lo, sc);
    __syncthreads();
    if (threadIdx.x < 10) atomicAdd(&hist[threadIdx.x], sh[threadIdx.x]);
}

__global__ void wg_kernel(const int* __restrict__ hist, int n, float* __restrict__ wg) {
    if (threadIdx.x == 0 && blockIdx.x == 0) {
        float Hb = entropy10(hist, (float)n);
        float Hg = entropy10(hist + 10, (float)n);
        *wg = 0.4f / 3.14159265358979f * atanf(500.f * (Hb - Hg)) + 0.5f;
    }
}

__global__ __launch_bounds__(256) void blend_kernel(const float* __restrict__ base,
                                                    const float* __restrict__ graft,
                                                    float* __restrict__ dw,
                                                    const float* __restrict__ wg, int n) {
    int i = blockIdx.x * blockDim.x + threadIdx.x;
    if (i < n) dw[i] = blend1(base[i], graft[i], dw[i], *wg);
}

__global__ void init_out_kernel(float* __restrict__ out, const float* __restrict__ b1, int n) {
    int i = blockIdx.x * blockDim.x + threadIdx.x;
    if (i < n) out[i] = b1[0];
}

// ---------------- fused layer-1 weight blend (W1: [1,512], p1: [512,512]) ----
__global__ __launch_bounds__(512) void layer1_kernel(const float* __restrict__ w1,
                                                     const float* __restrict__ g1,
                                                     const float* __restrict__ p1,
                                                     float* __restrict__ W1b) {
    __shared__ float ad[512];
    __shared__ unsigned red[4][512];
    __shared__ int hh[20];
    __shared__ float swg;
    int t = threadIdx.x;
    float wv = w1[t], gv = g1[t];
    ad[t] = fabsf(wv - gv);
    red[0][t] = fenc(wv); red[1][t] = fenc(wv);
    red[2][t] = fenc(gv); red[3][t] = fenc(gv);
    if (t < 20) hh[t] = 0;
    __syncthreads();
    for (int off = 256; off > 0; off >>= 1) {
        if (t < off) {
            red[0][t] = min(red[0][t], red[0][t + off]);
            red[1][t] = max(red[1][t], red[1][t + off]);
            red[2][t] = min(red[2][t], red[2][t + off]);
            red[3][t] = max(red[3][t], red[3][t + off]);
        }
        __syncthreads();
    }
    float wlo = fdec(red[0][0]), whi = fdec(red[1][0]);
    float glo = fdec(red[2][0]), ghi = fdec(red[3][0]);
    bin_add(hh,      wv, wlo, (whi - wlo) / 10.f);
    bin_add(hh + 10, gv, glo, (ghi - glo) / 10.f);
    __syncthreads();
    if (t == 0) {
        float Hb = entropy10(hh, 512.f);
        float Hg = entropy10(hh + 10, 512.f);
        swg = 0.4f / 3.14159265358979f * atanf(500.f * (Hb - Hg)) + 0.5f;
    }
    __syncthreads();
    float d = 0.f;
    const float* prow = p1 + (size_t)t * 512;
    #pragma unroll 8
    for (int k = 0; k < 512; ++k) d += ad[k] * prow[k];
    W1b[t] = blend1(wv, gv, d, swg);
}

// ---------------- pnet GEMM: D = |A - A2| @ B^T (fp32 WMMA, sync staging) ----
__global__ __launch_bounds__(256) void pnet_gemm_wmma(
    const float* __restrict__ A, const float* __restrict__ A2,
    const float* __restrict__ B, int K,
    float* __restrict__ D, int ldd) {

    __shared__ float xs[TM][LDSP];
    __shared__ float bs[TN][LDSP];

    const int t    = threadIdx.x;
    const int lane = t & 31;
    const int wid  = t >> 5;
    const int m0   = blockIdx.y * TM;
    const int n0   = blockIdx.x * TN;

    v8f acc[2][2] = {};

    const int wm = (wid & 3) * 32;
    const int wn = (wid >> 2) * 32;
    const int fm = lane & 15;
    const int fk = (lane >> 4) * 2;
    const int c4    = (t & 7) * 4;
    const int rbase = t >> 3;

    for (int k0 = 0; k0 < K; k0 += TK) {
        #pragma unroll
        for (int r = 0; r < 4; ++r) {
            int row = rbase + r * 32;
            v4f v = *(const v4f*)(A  + (size_t)(m0 + row) * K + k0 + c4);
            v4f u = *(const v4f*)(A2 + (size_t)(m0 + row) * K + k0 + c4);
            v.x = fabsf(v.x - u.x); v.y = fabsf(v.y - u.y);
            v.z = fabsf(v.z - u.z); v.w = fabsf(v.w - u.w);
            *(v4f*)&xs[row][c4] = v;
        }
        #pragma unroll
        for (int r = 0; r < 2; ++r) {
            int row = rbase + r * 32;
            *(v4f*)&bs[row][c4] = *(const v4f*)(B + (size_t)(n0 + row) * K + k0 + c4);
        }
        __syncthreads();
        #pragma unroll
        for (int kk = 0; kk < TK; kk += 4) {
            v2f af[2], bf[2];
            #pragma unroll
            for (int i = 0; i < 2; ++i) af[i] = *(const v2f*)&xs[wm + i * 16 + fm][kk + fk];
            #pragma unroll
            for (int j = 0; j < 2; ++j) bf[j] = *(const v2f*)&bs[wn + j * 16 + fm][kk + fk];
            #pragma unroll
            for (int i = 0; i < 2; ++i)
                #pragma unroll
                for (int j = 0; j < 2; ++j)
                    acc[i][j] = __builtin_amdgcn_wmma_f32_16x16x4_f32(
                        false, af[i], false, bf[j], (short)0, acc[i][j], false, false);
        }
        __syncthreads();
    }

    const int nLane = lane & 15;
    const int rOff  = (lane >> 4) * 8;
    #pragma unroll
    for (int i = 0; i < 2; ++i)
        #pragma unroll
        for (int j = 0; j < 2; ++j)
            #pragma unroll
            for (int r = 0; r < 8; ++r)
                D[(size_t)(m0 + wm + i * 16 + rOff + r) * ldd + (n0 + wn + j * 16 + nLane)] =
                    acc[i][j][r];
}

// ---------------- main GEMM: ReLU(X @ W^T + b0) fused with layer 1 -----------
// 128x128 block tile, 32x64 wave tile (2x4 fp32 WMMA accumulators), K step 32,
// double-buffered LDS filled by GLOBAL_LOAD_ASYNC_TO_LDS_B128 (ASYNCcnt).
__global__ __launch_bounds__(256) void main_gemm_async(
    const float* __restrict__ X, const float* __restrict__ W, int K,
    const float* __restrict__ bias, const float* __restrict__ w1,
    float* __restrict__ out) {

    __shared__ float xs[2][TM2][LDSP];
    __shared__ float bs[2][TN2][LDSP];
    __shared__ float rowsum[TM2];
    __shared__ float biasS[TN2];
    __shared__ float w1S[TN2];

    const int t    = threadIdx.x;
    const int lane = t & 31;
    const int wid  = t >> 5;
    const int m0   = blockIdx.y * TM2;
    const int n0   = blockIdx.x * TN2;

    if (t < TM2) rowsum[t] = 0.f;
    if (t < TN2) { biasS[t] = bias[n0 + t]; w1S[t] = w1[n0 + t]; }

    const int c4    = (t & 7) * 4;      // column within K-tile (float4 granularity)
    const int rbase = t >> 3;           // 0..31

    // LDS byte addresses (flat-pointer low 32 bits == wave-relative LDS offset)
    const unsigned xsBase = (unsigned)(size_t)&xs[0][0][0];
    const unsigned bsBase = (unsigned)(size_t)&bs[0][0][0];
    const unsigned bufX = (unsigned)(TM2 * LDSP * 4);
    const unsigned bufB = (unsigned)(TN2 * LDSP * 4);

    unsigned xg[4], bg[4], xl[4], bl[4];
    #pragma unroll
    for (int r = 0; r < 4; ++r) {
        int row = rbase + r * 32;
        xg[r] = (unsigned)(((size_t)(m0 + row) * K + c4) * 4);
        bg[r] = (unsigned)(((size_t)(n0 + row) * K + c4) * 4);
        xl[r] = (unsigned)((row * LDSP + c4) * 4);
        bl[r] = (unsigned)((row * LDSP + c4) * 4);
    }

    // prologue: stage 0 -> buffer 0 (8 async instructions per wave)
    #pragma unroll
    for (int r = 0; r < 4; ++r) {
        async_ld_b128(xsBase + xl[r], X, xg[r]);
        async_ld_b128(bsBase + bl[r], W, bg[r]);
    }

    v8f acc[2][4] = {};
    const int wm = (wid & 3) * 32;      // 4 waves in M
    const int wn = (wid >> 2) * 64;     // 2 waves in N
    const int fm = lane & 15;
    const int fk = (lane >> 4) * 2;

    const int NK = K / TK;
    for (int k = 0; k < NK; ++k) {
        const int buf = k & 1;
        if (k + 1 < NK) {
            const unsigned kOff = (unsigned)((k + 1) * TK * 4);
            const unsigned nbX  = xsBase + (buf ? 0u : bufX);
            const unsigned nbB  = bsBase + (buf ? 0u : bufB);
            #pragma unroll
            for (int r = 0; r < 4; ++r) {
                async_ld_b128(nbX + xl[r], X, xg[r] + kOff);
                async_ld_b128(nbB + bl[r], W, bg[r] + kOff);
            }
            S_WAIT_ASYNCCNT_8();    // oldest 8 (stage k) complete; stage k+1 in flight
        } else {
            S_WAIT_ASYNCCNT_0();    // drain final stage
        }
        __syncthreads();            // stage-k data from all waves visible

        const float (*xsb)[LDSP] = xs[buf];
        const float (*bsb)[LDSP] = bs[buf];
        #pragma unroll
        for (int kk = 0; kk < TK; kk += 4) {
            v2f af[2], bf[4];
            #pragma unroll
            for (int i = 0; i < 2; ++i) af[i] = *(const v2f*)&xsb[wm + i * 16 + fm][kk + fk];
            #pragma unroll
            for (int j = 0; j < 4; ++j) bf[j] = *(const v2f*)&bsb[wn + j * 16 + fm][kk + fk];
            #pragma unroll
            for (int i = 0; i < 2; ++i)
                #pragma unroll
                for (int j = 0; j < 4; ++j)
                    acc[i][j] = __builtin_amdgcn_wmma_f32_16x16x4_f32(
                        false, af[i], false, bf[j], (short)0, acc[i][j], false, false);
        }
        __syncthreads();            // all waves done reading buf before it is refilled
    }

    // Epilogue: ReLU + bias, dot with W1, LDS-reduce per row, one global atomic/row.
    const int nLane = lane & 15;
    const int rOff  = (lane >> 4) * 8;
    #pragma unroll
    for (int i = 0; i < 2; ++i) {
        #pragma unroll
        for (int r = 0; r < 8; ++r) {
            float s = 0.f;
            #pragma unroll
            for (int j = 0; j < 4; ++j) {
                int nl = wn + j * 16 + nLane;
                float h = fmaxf(acc[i][j][r] + biasS[nl], 0.f);   // layer-0 ReLU
                s += h * w1S[nl];                                 // layer-1 partial dot
            }
            atomicAdd(&rowsum[wm + i * 16 + rOff + r], s);        // ds_add_f32
        }
    }
    __syncthreads();
    if (t < TM2) atomicAdd(&out[m0 + t], rowsum[t]);              // global_atomic_add_f32
}

// ---------------- launch -----------------------------------------------------
extern "C" void kernel_launch(void* const* d_in, const int* in_sizes, int n_in,
                              void* d_out, int out_size, void* d_ws, size_t ws_size,
                              hipStream_t stream) {
    const float* x  = (const float*)d_in[0];   // [65536,1024]
    const float* w0 = (const float*)d_in[1];   // [512,1024]
    const float* b0 = (const float*)d_in[2];   // [512]
    const float* w1 = (const float*)d_in[3];   // [1,512]
    const float* b1 = (const float*)d_in[4];   // [1]
    const float* p0 = (const float*)d_in[5];   // [1024,1024]
    const float* p1 = (const float*)d_in[6];   // [512,512]
    const float* g0 = (const float*)d_in[7];   // [512,1024]
    const float* g1 = (const float*)d_in[8];   // [1,512]
    float* out = (float*)d_out;                // [65536,1]

    float*    wsf  = (float*)d_ws;
    float*    W0b  = wsf;                          // 524288 floats (blended W0)
    float*    W1b  = wsf + 524288;                 // 512 floats (blended W1)
    unsigned* enc  = (unsigned*)(wsf + 524800);    // 4: [w0min,w0max,g0min,g0max]
    int*      hist = (int*)(wsf + 524804);         // 20: hist(w0) | hist(g0)
    float*    wg0  = wsf + 524824;                 // 1: w_global for layer 0

    const int NW0 = 512 * 1024;

    init_stats_kernel<<<1, 32, 0, stream>>>(enc, hist);
    minmax2_kernel<<<256, 256, 0, stream>>>(w0, g0, NW0, enc);
    hist_kernel<<<256, 256, 0, stream>>>(w0, NW0, enc + 0, hist + 0);
    hist_kernel<<<256, 256, 0, stream>>>(g0, NW0, enc + 2, hist + 10);
    wg_kernel<<<1, 1, 0, stream>>>(hist, NW0, wg0);

    // d0 = |w0 - g0| @ p0^T   (M=512, N=1024, K=1024) -> raw scores into W0b
    pnet_gemm_wmma<<<dim3(1024 / TN, 512 / TM), 256, 0, stream>>>(
        w0, g0, p0, 1024, W0b, 1024);

    // W0b = blend(w0, g0, sigmoid(d0), wg0)  (in place)
    blend_kernel<<<(NW0 + 255) / 256, 256, 0, stream>>>(w0, g0, W0b, wg0, NW0);

    // Entire W1 blend path (entropy + pnet matvec + blend) in one small kernel.
    layer1_kernel<<<1, 512, 0, stream>>>(w1, g1, p1, W1b);

    // out = b1 broadcast; main GEMM atomically accumulates layer-1 dot products.
    init_out_kernel<<<(65536 + 255) / 256, 256, 0, stream>>>(out, b1, 65536);

    // ReLU(x @ W0b^T + b0) fused with layer 1 (M=65536, N=512, K=1024)
    main_gemm_async<<<dim3(512 / TN2, 65536 / TM2), 256, 0, stream>>>(
        x, W0b, 1024, b0, W1b, out);
}